// VectorQuantizerEMA_16406775071544
// MI455X (gfx1250) — compile-verified
//
#include <hip/hip_runtime.h>
#include <hip/hip_bf16.h>
#include <math.h>

typedef __attribute__((ext_vector_type(2))) float v2f;
typedef __attribute__((ext_vector_type(8))) float v8f;

#define NUM_EMB 8
#define EMB_DIM 64
#define HW      16384          /* 128*128 */
#define N_TOK   1048576        /* 64*HW tokens */
#define TOTAL   67108864       /* 64*64*HW elements */
#define DECAYF  0.99f
#define COMMITF 0.25f
#define EPSF    1e-5f

/* workspace layout in floats */
#define WS_CNT  0      /* 8  : counts            */
#define WS_DW   8      /* 512: dw[8][64]         */
#define WS_NW   520    /* 512: new_weight[8][64] */
#define WS_LOSS 1032   /* 1  : sum-sq accum      */
#define WS_TOTF 1040

/* ds_swizzle xor-mode: offset[14:10]=xor_mask, [4:0]=and_mask(0x1f) */
template <int XORM>
__device__ __forceinline__ unsigned swz_xor_u(unsigned v) {
    return (unsigned)__builtin_amdgcn_ds_swizzle((int)v, (XORM << 10) | 0x1f);
}
template <int XORM>
__device__ __forceinline__ float swz_xor_f(float v) {
    return __int_as_float(
        __builtin_amdgcn_ds_swizzle(__float_as_int(v), (XORM << 10) | 0x1f));
}

__global__ void k_init(float* __restrict__ ws) {
    for (int i = threadIdx.x; i < WS_TOTF; i += blockDim.x) ws[i] = 0.0f;
}

/* ---------------- pass 1: assignment (WMMA distances) + counts/dw ---------------- */
__global__ void k_assign(const float* __restrict__ x, const float* __restrict__ emb,
                         float* __restrict__ ws, float* __restrict__ idxF) {
    __shared__ float dwL[NUM_EMB * EMB_DIM];
    __shared__ float cntL[NUM_EMB];
    __shared__ int   tIdxL[8 * 16];

    const int t = threadIdx.x;
    for (int i = t; i < NUM_EMB * EMB_DIM; i += blockDim.x) dwL[i] = 0.0f;
    if (t < NUM_EMB) cntL[t] = 0.0f;
    __syncthreads();

    const int lane = t & 31;
    const int wave = t >> 5;
    const int n    = lane & 15;   /* code column N / token row M within half */
    const int h    = lane >> 4;   /* half-wave selects K pair                */

    /* B fragments (4x16 f32 per step): lane n+16h holds B[2h][n], B[2h+1][n] */
    v2f Bf[16];
    float part = 0.0f;
#pragma unroll
    for (int s = 0; s < 16; ++s) {
        const int c0 = 4 * s + 2 * h;
        float bx = (n < NUM_EMB) ? emb[n * EMB_DIM + c0]     : 0.0f;
        float by = (n < NUM_EMB) ? emb[n * EMB_DIM + c0 + 1] : 0.0f;
        Bf[s].x = bx; Bf[s].y = by;
        part += bx * bx + by * by;
    }
    float e2 = part + swz_xor_f<16>(part);        /* full ||e_n||^2 */
    if (n >= NUM_EMB) e2 = 3.0e38f;               /* padded columns never win */

    const int nTiles = N_TOK / 16;
    const int gWaves = (gridDim.x * blockDim.x) >> 5;
    const int gw     = (blockIdx.x * blockDim.x + t) >> 5;   /* wave-uniform */

    for (int tile = gw; tile < nTiles; tile += gWaves) {
        const long tokenBase = (long)tile * 16;
        const int  b  = (int)(tokenBase >> 14);
        const int  hw = (int)(tokenBase & 16383);
        const float* xb = x + (long)b * (long)(EMB_DIM * HW) + hw;

        /* dots = X(16x64) * E^T(64x16) via 16 chained WMMA f32 16x16x4.
           Keep A fragments live: they also feed the dw accumulation below. */
        v2f Af[16];
        v8f acc = {0.f, 0.f, 0.f, 0.f, 0.f, 0.f, 0.f, 0.f};
#pragma unroll
        for (int s = 0; s < 16; ++s) {
            const int c0 = 4 * s + 2 * h;
            v2f a;
            a.x = xb[(long)c0 * HW + n];         /* A[m=n][K=2h]   */
            a.y = xb[(long)(c0 + 1) * HW + n];   /* A[m=n][K=2h+1] */
            Af[s] = a;
            acc = __builtin_amdgcn_wmma_f32_16x16x4_f32(
                      false, a, false, Bf[s], (short)0, acc, false, false);
        }

        /* per C register r: token M = r + 8h, N = lane%16.
           Order-preserving uint key with index in low 4 bits; butterfly v_min_u32;
           every lane then holds the winner -> ki = key & 15 (no ballot needed). */
#pragma unroll
        for (int r = 0; r < 8; ++r) {
            const float val = e2 - 2.0f * acc[r];
            const int bits = __float_as_int(val);
            unsigned key = (unsigned)bits ^ (((unsigned)(bits >> 31)) | 0x80000000u);
            key = (key & 0xFFFFFFF0u) | (unsigned)n;
            key = min(key, swz_xor_u<1>(key));
            key = min(key, swz_xor_u<2>(key));
            key = min(key, swz_xor_u<4>(key));
            key = min(key, swz_xor_u<8>(key));
            const int ki = (int)(key & 15u);
            if (n == 0) {   /* lanes 0 and 16 publish their half's token */
                tIdxL[wave * 16 + 8 * h + r] = ki;
                idxF[tokenBase + 8 * h + r]  = (float)ki;
            }
        }

        /* counts + dw straight from the A fragments:
           lane (n,h) owns token n channels {4s+2h, 4s+2h+1}; halves cover all 64. */
        const int k = tIdxL[wave * 16 + n];
#pragma unroll
        for (int s = 0; s < 16; ++s) {
            const int c0 = 4 * s + 2 * h;
            atomicAdd(&dwL[k * EMB_DIM + c0],     Af[s].x);
            atomicAdd(&dwL[k * EMB_DIM + c0 + 1], Af[s].y);
        }
        if (h == 0) atomicAdd(&cntL[k], 1.0f);
    }

    __syncthreads();
    for (int i = t; i < NUM_EMB * EMB_DIM; i += blockDim.x) atomicAdd(&ws[WS_DW + i], dwL[i]);
    if (t < NUM_EMB) atomicAdd(&ws[WS_CNT + t], cntL[t]);
}

/* ---------------- pass 2: EMA codebook update + perplexity (tiny) ---------------- */
__global__ void k_update(const float* __restrict__ ecs, const float* __restrict__ emaw,
                         float* __restrict__ ws, float* __restrict__ perpOut) {
    __shared__ float cs[NUM_EMB];
    __shared__ float nsh;
    const int d = threadIdx.x;   /* 64 threads */
    if (d < NUM_EMB) cs[d] = ecs[d] * DECAYF + (1.0f - DECAYF) * ws[WS_CNT + d];
    __syncthreads();
    if (d == 0) { float s = 0.f; for (int k = 0; k < NUM_EMB; ++k) s += cs[k]; nsh = s; }
    __syncthreads();
    const float nn = nsh;
    __syncthreads();
    if (d < NUM_EMB) cs[d] = (cs[d] + EPSF) / (nn + NUM_EMB * EPSF) * nn;
    __syncthreads();
    for (int k = 0; k < NUM_EMB; ++k) {
        const float nw = emaw[k * EMB_DIM + d] * DECAYF
                       + (1.0f - DECAYF) * ws[WS_DW + k * EMB_DIM + d];
        ws[WS_NW + k * EMB_DIM + d] = nw / cs[k];
    }
    if (d == 0) {
        float s = 0.0f;
        for (int k = 0; k < NUM_EMB; ++k) {
            const float p = ws[WS_CNT + k] / (float)N_TOK;
            s += p * logf(p + 1e-10f);
        }
        perpOut[0] = expf(-s);
    }
}

/* ---------------- pass 3: quantize (new weights), write NCHW q, MSE ---------------- */
__global__ void k_quant(const float* __restrict__ x, const float* __restrict__ ws,
                        const float* __restrict__ idxF, float* __restrict__ qOut,
                        float* __restrict__ lossAcc) {
    __shared__ float nw[NUM_EMB * EMB_DIM];
    __shared__ int   kL[256];
    __shared__ float lred;
    const int t = threadIdx.x;
    for (int i = t; i < NUM_EMB * EMB_DIM; i += 256) nw[i] = ws[WS_NW + i];
    const long token = (long)blockIdx.x * 256 + t;
    kL[t] = (int)idxF[token];
    if (t == 0) lred = 0.0f;
    __syncthreads();

    const int b  = (int)(token >> 14);
    const int hw = (int)(token & 16383);
    const float* xb = x    + (long)b * (long)(EMB_DIM * HW) + hw;
    float*       qb = qOut + (long)b * (long)(EMB_DIM * HW) + hw;
    const float* nwr = &nw[kL[t] * EMB_DIM];

    float ls = 0.0f;
#pragma unroll 8
    for (int c = 0; c < EMB_DIM; ++c) {
        const float q  = nwr[c];
        const float xv = xb[(long)c * HW];
        qb[(long)c * HW] = q;                 /* q_st value == quantized */
        const float dd = q - xv;
        ls += dd * dd;
    }
    /* wave32 sum via swizzle butterflies, then one LDS + one global atomic per block */
    ls += swz_xor_f<1>(ls);
    ls += swz_xor_f<2>(ls);
    ls += swz_xor_f<4>(ls);
    ls += swz_xor_f<8>(ls);
    ls += swz_xor_f<16>(ls);
    if ((t & 31) == 0) atomicAdd(&lred, ls);
    __syncthreads();
    if (t == 0) atomicAdd(lossAcc, lred);
}

__global__ void k_fin(const float* __restrict__ ws, float* __restrict__ loss) {
    loss[0] = COMMITF * ws[WS_LOSS] / (float)TOTAL;
}

extern "C" void kernel_launch(void* const* d_in, const int* in_sizes, int n_in,
                              void* d_out, int out_size, void* d_ws, size_t ws_size,
                              hipStream_t stream) {
    const float* x    = (const float*)d_in[0];   /* (64,64,128,128) NCHW  */
    const float* emb  = (const float*)d_in[1];   /* (8,64)                */
    const float* ecs  = (const float*)d_in[2];   /* (8,)                  */
    const float* emaw = (const float*)d_in[3];   /* (8,64)                */
    float* out = (float*)d_out;
    float* ws  = (float*)d_ws;

    float* lossOut = out;                  /* [0]                 */
    float* qOut    = out + 1;              /* [1 .. TOTAL]        */
    float* perpOut = out + 1 + TOTAL;      /* [TOTAL+1]           */
    float* idxOut  = out + 2 + TOTAL;      /* [TOTAL+2 ..] 1M idx */

    k_init  <<<1,            256, 0, stream>>>(ws);
    k_assign<<<1024,         256, 0, stream>>>(x, emb, ws, idxOut);
    k_update<<<1,            64,  0, stream>>>(ecs, emaw, ws, perpOut);
    k_quant <<<N_TOK / 256,  256, 0, stream>>>(x, ws, idxOut, qOut, ws + WS_LOSS);
    k_fin   <<<1,            1,   0, stream>>>(ws, lossOut);
}